// Node_Layer_14499809591359
// MI455X (gfx1250) — compile-verified
//
#include <hip/hip_runtime.h>

typedef float v2f __attribute__((ext_vector_type(2)));
typedef float v8f __attribute__((ext_vector_type(8)));

#define N_NODES 50000
#define N_EDGES 640000
#define D_NODE 128
#define D_EDGE 128
#define IN_NF 256   // D_NODE + D_EDGE
#define H_NF 256
#define OUT_NF 128

#define W1_ELEMS (IN_NF * H_NF)    // 65536
#define W2_ELEMS (H_NF * OUT_NF)   // 32768

#define ROWS_PER_BLOCK 32

// ---------------------------------------------------------------- zero d_out
__global__ void zero_out_kernel(float* __restrict__ out, int n4) {
    int i = blockIdx.x * blockDim.x + threadIdx.x;
    if (i < n4) ((float4*)out)[i] = make_float4(0.f, 0.f, 0.f, 0.f);
}

// ---------------------------------------- transpose W1, W2 into workspace
// W1t[n*IN_NF + k] = W1[k*H_NF + n]   (256x256)
// W2t[n*H_NF  + k] = W2[k*OUT_NF + n] (128x256)
__global__ void transpose_w_kernel(const float* __restrict__ W1,
                                   const float* __restrict__ W2,
                                   float* __restrict__ W1t,
                                   float* __restrict__ W2t) {
    int idx = blockIdx.x * blockDim.x + threadIdx.x;
    if (idx < W1_ELEMS) {
        int k = idx >> 8;          // H_NF == 256 columns in W1 row-major
        int n = idx & 255;
        W1t[n * IN_NF + k] = W1[idx];
    }
    if (idx < W2_ELEMS) {
        int k = idx >> 7;          // OUT_NF == 128 columns in W2 row-major
        int n = idx & 127;
        W2t[n * H_NF + k] = W2[idx];
    }
}

// ------------------------------------------- scatter-add edge_attr -> agg(out)
// one thread handles 4 consecutive floats of one edge (float4 load, 4 atomics)
__global__ void scatter_add_kernel(const float* __restrict__ edge_attr,
                                   const int* __restrict__ edge_index,
                                   float* __restrict__ agg) {
    int idx = blockIdx.x * blockDim.x + threadIdx.x;   // N_EDGES * 32 threads
    int e = idx >> 5;
    if (e >= N_EDGES) return;
    int c = (idx & 31) << 2;
    int row = edge_index[e];                            // edge_index[0][e]
    float4 v = *(const float4*)(edge_attr + (size_t)e * D_EDGE + c);
    float* dst = agg + (size_t)row * D_EDGE + c;
    atomicAdd(dst + 0, v.x);
    atomicAdd(dst + 1, v.y);
    atomicAdd(dst + 2, v.z);
    atomicAdd(dst + 3, v.w);
}

// --------------------------------------------- fused 2-layer MLP, fp32 WMMA
// block = 256 threads (8 wave32), 32 node rows per block (two 16-row halves
// sharing every B-fragment load -> 2 WMMAs per global_load_b64).
// One 32x256 LDS buffer is reused: node_in tile, then (after barrier) h tile.
__global__ __launch_bounds__(256) void fused_mlp_kernel(
    const float* __restrict__ node_feats,
    const float* __restrict__ W1t, const float* __restrict__ b1,
    const float* __restrict__ W2t, const float* __restrict__ b2,
    float* __restrict__ out) {

    __shared__ float lds[ROWS_PER_BLOCK * IN_NF];   // 32 KB

    const int tid  = threadIdx.x;
    const int wave = tid >> 5;
    const int lane = tid & 31;
    const int half = lane >> 4;   // 0: lanes 0-15, 1: lanes 16-31
    const int mrow = lane & 15;
    const int m0   = blockIdx.x * ROWS_PER_BLOCK;
    const bool hi_valid = (m0 + 16) < N_NODES;      // block-uniform

    // ---- stage node_in tile into LDS (cols 0..127 node_feats, 128..255 agg)
    for (int i = tid; i < ROWS_PER_BLOCK * IN_NF; i += 256) {
        int r   = i >> 8;      // IN_NF == 256
        int col = i & 255;
        int row = m0 + r;
        if (row >= N_NODES) row = N_NODES - 1;      // clamp (tail block)
        float v = (col < D_NODE)
                      ? node_feats[(size_t)row * D_NODE + col]
                      : out[(size_t)row * D_EDGE + (col - D_NODE)];
        lds[i] = v;
    }
    __syncthreads();

    // ---- stage 1: h = node_in @ W1 + b1 ; wave owns cols [wave*32, wave*32+32)
    const int n0 = wave * 32;
    const int n1 = n0 + 16;
    v8f c00 = {};   // rows 0..15,  cols n0..n0+15
    v8f c01 = {};   // rows 0..15,  cols n1..n1+15
    v8f c10 = {};   // rows 16..31, cols n0..n0+15
    v8f c11 = {};   // rows 16..31, cols n1..n1+15
    {
        const float* __restrict__ wrow0 = W1t + (size_t)(n0 + mrow) * IN_NF;
        const float* __restrict__ wrow1 = W1t + (size_t)(n1 + mrow) * IN_NF;
        for (int k0 = 0; k0 < IN_NF; k0 += 4) {
            const int ka = k0 + 2 * half;           // ISA 16x4 f32 A layout
            v2f a_lo = *(const v2f*)&lds[ mrow       * IN_NF + ka];
            v2f a_hi = *(const v2f*)&lds[(mrow + 16) * IN_NF + ka];
            v2f bf0  = *(const v2f*)&wrow0[ka];
            v2f bf1  = *(const v2f*)&wrow1[ka];
            c00 = __builtin_amdgcn_wmma_f32_16x16x4_f32(false, a_lo, false, bf0,
                                                        (short)0, c00, false, false);
            c10 = __builtin_amdgcn_wmma_f32_16x16x4_f32(false, a_hi, false, bf0,
                                                        (short)0, c10, false, false);
            c01 = __builtin_amdgcn_wmma_f32_16x16x4_f32(false, a_lo, false, bf1,
                                                        (short)0, c01, false, false);
            c11 = __builtin_amdgcn_wmma_f32_16x16x4_f32(false, a_hi, false, bf1,
                                                        (short)0, c11, false, false);
        }
    }
    __syncthreads();   // all stage-1 reads of lds complete

    // ---- write h (+bias) into the same LDS buffer
    {
        const float bias0 = b1[n0 + mrow];
        const float bias1 = b1[n1 + mrow];
#pragma unroll
        for (int i = 0; i < 8; ++i) {
            const int M = i + 8 * half;             // C/D VGPR striping
            lds[ M       * H_NF + n0 + mrow] = c00[i] + bias0;
            lds[ M       * H_NF + n1 + mrow] = c01[i] + bias1;
            lds[(M + 16) * H_NF + n0 + mrow] = c10[i] + bias0;
            lds[(M + 16) * H_NF + n1 + mrow] = c11[i] + bias1;
        }
    }
    __syncthreads();

    // ---- stage 2: out = h @ W2 + b2 ; wave owns cols [wave*16, wave*16+16)
    {
        const int on0 = wave * 16;
        const float* __restrict__ wrow = W2t + (size_t)(on0 + mrow) * H_NF;
        v8f d0 = {};   // rows 0..15
        v8f d1 = {};   // rows 16..31
        for (int k0 = 0; k0 < H_NF; k0 += 4) {
            const int ka = k0 + 2 * half;
            v2f a_lo = *(const v2f*)&lds[ mrow       * H_NF + ka];
            v2f a_hi = *(const v2f*)&lds[(mrow + 16) * H_NF + ka];
            v2f bf   = *(const v2f*)&wrow[ka];
            d0 = __builtin_amdgcn_wmma_f32_16x16x4_f32(false, a_lo, false, bf,
                                                       (short)0, d0, false, false);
            d1 = __builtin_amdgcn_wmma_f32_16x16x4_f32(false, a_hi, false, bf,
                                                       (short)0, d1, false, false);
        }
        const float bias = b2[on0 + mrow];
#pragma unroll
        for (int i = 0; i < 8; ++i) {
            const int M = i + 8 * half;
            out[(size_t)(m0 + M) * OUT_NF + on0 + mrow] = d0[i] + bias;
        }
        if (hi_valid) {
#pragma unroll
            for (int i = 0; i < 8; ++i) {
                const int M = i + 8 * half;
                out[(size_t)(m0 + 16 + M) * OUT_NF + on0 + mrow] = d1[i] + bias;
            }
        }
    }
}

extern "C" void kernel_launch(void* const* d_in, const int* in_sizes, int n_in,
                              void* d_out, int out_size, void* d_ws, size_t ws_size,
                              hipStream_t stream) {
    const float* node_feats = (const float*)d_in[0];
    const int*   edge_index = (const int*)  d_in[1];  // JAX default x64-off: int32
    const float* edge_attr  = (const float*)d_in[2];
    const float* W1         = (const float*)d_in[3];
    const float* b1         = (const float*)d_in[4];
    const float* W2         = (const float*)d_in[5];
    const float* b2         = (const float*)d_in[6];
    float* out = (float*)d_out;

    float* W1t = (float*)d_ws;              // 256 KB
    float* W2t = W1t + W1_ELEMS;            // 128 KB  (total 384 KB of ws)

    // 1) zero agg region (aliased onto d_out)
    const int n4 = N_NODES * OUT_NF / 4;
    zero_out_kernel<<<(n4 + 255) / 256, 256, 0, stream>>>(out, n4);

    // 2) transpose weights into workspace (B-fragments become b64 loads)
    transpose_w_kernel<<<(W1_ELEMS + 255) / 256, 256, 0, stream>>>(
        W1, W2, W1t, W2t);

    // 3) scatter-add edge features onto source nodes
    const int sthreads = N_EDGES * 32;
    scatter_add_kernel<<<(sthreads + 255) / 256, 256, 0, stream>>>(
        edge_attr, edge_index, out);

    // 4) fused MLP: concat -> GEMM1(+b1) -> GEMM2(+b2), in-place over agg rows
    const int nblocks = (N_NODES + ROWS_PER_BLOCK - 1) / ROWS_PER_BLOCK;
    fused_mlp_kernel<<<nblocks, 256, 0, stream>>>(
        node_feats, W1t, b1, W2t, b2, out);
}